// QuantumCircuit_71408126263792
// MI455X (gfx1250) — compile-verified
//
#include <hip/hip_runtime.h>
#include <hip/hip_bf16.h>

typedef __attribute__((ext_vector_type(2))) float v2f;
typedef __attribute__((ext_vector_type(8))) float v8f;

// ---------------------------------------------------------------------------
// 2x2 / 4x4 factor entries (indexed by row/col bits)
// ---------------------------------------------------------------------------
__device__ __forceinline__ float ryE(float cc, float ss, int i, int j) {
    // RY = [[c, -s], [s, c]]
    return (i == 0) ? ((j == 0) ? cc : -ss) : ((j == 0) ? ss : cc);
}
__device__ __forceinline__ float xE(int i, int j) { return (i != j) ? 1.f : 0.f; }
__device__ __forceinline__ float iE(int i, int j) { return (i == j) ? 1.f : 0.f; }
__device__ __forceinline__ float cnotE(int i, int j) {
    // rows map 0->0, 1->1, 2->3, 3->2
    int tgt = (i < 2) ? i : (i ^ 1);
    return (j == tgt) ? 1.f : 0.f;
}

// ---------------------------------------------------------------------------
// Kernel 1: build the 16x16 circuit unitary U from the 6 thetas (1 block/256 thr)
// ---------------------------------------------------------------------------
__global__ void __launch_bounds__(256) build_u_kernel(
    const float* __restrict__ th1, const float* __restrict__ th2,
    const float* __restrict__ th3, const float* __restrict__ th4,
    const float* __restrict__ th5, const float* __restrict__ th6,
    float* __restrict__ Uout) {
    __shared__ float S0[256], S1[256], S2[256];
    const int t = threadIdx.x;
    const int r = t >> 4, c = t & 15;
    const int r3 = (r >> 3) & 1, r2 = (r >> 2) & 1, r1 = (r >> 1) & 1, r0 = r & 1;
    const int c3 = (c >> 3) & 1, c2 = (c >> 2) & 1, c1 = (c >> 1) & 1, c0 = c & 1;

    const float h1 = th1[0] * 0.5f, h2 = th2[0] * 0.5f, h3 = th3[0] * 0.5f;
    const float h4 = th4[0] * 0.5f, h5 = th5[0] * 0.5f, h6 = th6[0] * 0.5f;
    const float c1_ = cosf(h1), s1_ = sinf(h1);
    const float c2_ = cosf(h2), s2_ = sinf(h2);
    const float c3_ = cosf(h3), s3_ = sinf(h3);
    const float c4_ = cosf(h4), s4_ = sinf(h4);
    const float c5_ = cosf(h5), s5_ = sinf(h5);
    const float c6_ = cosf(h6), s6_ = sinf(h6);

    // layer1 = X (x) X (x) RY(t1) (x) RY(t2)
    S0[t] = xE(r3, c3) * xE(r2, c2) * ryE(c1_, s1_, r1, c1) * ryE(c2_, s2_, r0, c0);
    // layer2 = RY(t3) (x) RY(t4) (x) CNOT
    S1[t] = ryE(c3_, s3_, r3, c3) * ryE(c4_, s4_, r2, c2) * cnotE(r1 * 2 + r0, c1 * 2 + c0);
    __syncthreads();

    float acc = 0.f;
    #pragma unroll
    for (int k = 0; k < 16; ++k) acc += S1[r * 16 + k] * S0[k * 16 + c];
    S2[t] = acc;  // T1 = L2 @ L1
    __syncthreads();

    // layer3 = CNOT (x) I (x) RY(t5)
    S0[t] = cnotE(r3 * 2 + r2, c3 * 2 + c2) * iE(r1, c1) * ryE(c5_, s5_, r0, c0);
    __syncthreads();
    acc = 0.f;
    #pragma unroll
    for (int k = 0; k < 16; ++k) acc += S0[r * 16 + k] * S2[k * 16 + c];
    S1[t] = acc;  // T2 = L3 @ T1
    __syncthreads();

    // layer4 = I (x) RY(t6) (x) I (x) I
    S0[t] = iE(r3, c3) * ryE(c6_, s6_, r2, c2) * iE(r1, c1) * iE(r0, c0);
    __syncthreads();
    acc = 0.f;
    #pragma unroll
    for (int k = 0; k < 16; ++k) acc += S0[r * 16 + k] * S1[k * 16 + c];
    S2[t] = acc;  // T3 = L4 @ T2
    __syncthreads();

    // layer5: kron(I,I,CNOT) with row axes (1,2) and col axes (5,6) swapped:
    // L5[r,c] = d(r3,c3) * d(r1,c1) * CNOT[2*r2+r0, 2*c2+c0]
    S0[t] = iE(r3, c3) * iE(r1, c1) * cnotE(r2 * 2 + r0, c2 * 2 + c0);
    __syncthreads();
    acc = 0.f;
    #pragma unroll
    for (int k = 0; k < 16; ++k) acc += S0[r * 16 + k] * S2[k * 16 + c];
    Uout[t] = acc;  // U = L5 @ T3
}

// ---------------------------------------------------------------------------
// Kernel 2: alpha = U @ input_state   (16 x Bcols), one column per thread
// ---------------------------------------------------------------------------
__global__ void __launch_bounds__(256) compute_alpha_kernel(
    const float* __restrict__ U, const float* __restrict__ inp,
    float* __restrict__ alpha, int Bcols) {
    __shared__ float Us[256];
    const int t = threadIdx.x;
    Us[t] = U[t];
    __syncthreads();
    const int c = blockIdx.x * 256 + t;
    float x[16];
    #pragma unroll
    for (int k = 0; k < 16; ++k) x[k] = inp[(size_t)k * Bcols + c];
    #pragma unroll
    for (int row = 0; row < 16; ++row) {
        float s = 0.f;
        #pragma unroll
        for (int k = 0; k < 16; ++k) s += Us[row * 16 + k] * x[k];
        alpha[(size_t)row * Bcols + c] = s;
    }
}

// ---------------------------------------------------------------------------
// Kernel 3: Out = alpha^T * diag(+1,-1,...) * alpha   via v_wmma_f32_16x16x4_f32
// Block = 256 thr (8 waves) -> 128(M) x 64(N) tile; each wave 32x32.
// ---------------------------------------------------------------------------
__global__ void __launch_bounds__(256) gemm_rank16_kernel(
    const float* __restrict__ alpha, float* __restrict__ out, int Bcols) {
    const int lane = threadIdx.x & 31;
    const int wave = threadIdx.x >> 5;
    const int lm = lane & 15;            // M / N index within 16-wide fragment
    const int lkoff = (lane >> 4) << 1;  // K offset 0 or 2 (A/B fp32 frag layout)
    const int m0 = blockIdx.y * 128 + (wave >> 1) * 32;
    const int n0 = blockIdx.x * 64 + (wave & 1) * 32;

    v8f acc[2][2] = {};

    #pragma unroll
    for (int kc = 0; kc < 4; ++kc) {
        const int kb = (kc << 2) + lkoff;  // always even -> sign +1
        const float* __restrict__ re = alpha + (size_t)kb * Bcols;        // even k row
        const float* __restrict__ ro = alpha + (size_t)(kb + 1) * Bcols;  // odd  k row
        v2f afrag[2], bfrag[2];
        #pragma unroll
        for (int i = 0; i < 2; ++i) {
            afrag[i][0] = re[m0 + i * 16 + lm];
            afrag[i][1] = ro[m0 + i * 16 + lm];
            bfrag[i][0] = re[n0 + i * 16 + lm];   // s_k = +1 (even k)
            bfrag[i][1] = -ro[n0 + i * 16 + lm];  // s_k = -1 (odd k): fold diag(M)
        }
        #pragma unroll
        for (int mi = 0; mi < 2; ++mi)
            #pragma unroll
            for (int ni = 0; ni < 2; ++ni)
                acc[mi][ni] = __builtin_amdgcn_wmma_f32_16x16x4_f32(
                    false, afrag[mi], false, bfrag[ni], (short)0, acc[mi][ni],
                    false, false);
    }

    // D layout: VGPR r -> row r (lanes 0-15) / row r+8 (lanes 16-31), col = lane&15
    const int rowHi = (lane >> 4) << 3;  // 0 or 8
    #pragma unroll
    for (int mi = 0; mi < 2; ++mi)
        #pragma unroll
        for (int ni = 0; ni < 2; ++ni)
            #pragma unroll
            for (int rr = 0; rr < 8; ++rr) {
                size_t row = (size_t)(m0 + mi * 16 + rr + rowHi);
                size_t col = (size_t)(n0 + ni * 16 + lm);
                __builtin_nontemporal_store(acc[mi][ni][rr],
                                            out + row * (size_t)Bcols + col);
            }
}

// ---------------------------------------------------------------------------
extern "C" void kernel_launch(void* const* d_in, const int* in_sizes, int n_in,
                              void* d_out, int out_size, void* d_ws, size_t ws_size,
                              hipStream_t stream) {
    (void)n_in; (void)out_size; (void)ws_size;
    const float* inp = (const float*)d_in[0];
    const float* th1 = (const float*)d_in[1];
    const float* th2 = (const float*)d_in[2];
    const float* th3 = (const float*)d_in[3];
    const float* th4 = (const float*)d_in[4];
    const float* th5 = (const float*)d_in[5];
    const float* th6 = (const float*)d_in[6];
    float* outp = (float*)d_out;

    const int Bcols = in_sizes[0] / 16;  // 8192

    float* U = (float*)d_ws;             // 256 floats
    float* alpha = (float*)d_ws + 256;   // 16 * Bcols floats (512 KB)

    build_u_kernel<<<1, 256, 0, stream>>>(th1, th2, th3, th4, th5, th6, U);
    compute_alpha_kernel<<<Bcols / 256, 256, 0, stream>>>(U, inp, alpha, Bcols);

    dim3 grid(Bcols / 64, Bcols / 128);  // (128, 64) blocks
    gemm_rank16_kernel<<<grid, 256, 0, stream>>>(alpha, outp, Bcols);
}